// MultiBoxLoss_6493990551569
// MI455X (gfx1250) — compile-verified
//
#include <hip/hip_runtime.h>
#include <hip/hip_bf16.h>
#include <stdint.h>

// ---------------------------------------------------------------------------
// MultiBoxLoss for gfx1250 (MI455X).
// Deterministic: only integer atomics (add/max, commutative+exact) cross
// blocks; all float reductions are fixed-order trees / WMMA (fixed HW order).
// ---------------------------------------------------------------------------

#define TMAX 64  // max truths we stage in LDS (reference uses T=48)

typedef float v2f __attribute__((ext_vector_type(2)));
typedef float v8f __attribute__((ext_vector_type(8)));

#if defined(__gfx1250__) && __has_builtin(__builtin_amdgcn_wmma_f32_16x16x4_f32)
#define USE_WMMA_REDUCE 1
#else
#define USE_WMMA_REDUCE 0
#endif

#if defined(__gfx1250__) && __has_builtin(__builtin_amdgcn_global_load_async_to_lds_b32) && \
    __has_builtin(__builtin_amdgcn_s_wait_asynccnt)
#define USE_ASYNC_LDS 1
#else
#define USE_ASYNC_LDS 0
#endif

__device__ __forceinline__ float smooth_l1(float d) {
  d = fabsf(d);
  return d < 1.0f ? 0.5f * d * d : d - 0.5f;
}

// Cooperative stage of `count` floats into LDS; caller must __syncthreads().
__device__ __forceinline__ void stage_to_lds(const float* __restrict__ gsrc,
                                             float* sdst, int count) {
#if USE_ASYNC_LDS
  for (int i = threadIdx.x; i < count; i += blockDim.x) {
    __builtin_amdgcn_global_load_async_to_lds_b32(
        (__attribute__((address_space(1))) int*)(gsrc + i),
        (__attribute__((address_space(3))) int*)(sdst + i),
        0, 0);
  }
  __builtin_amdgcn_s_wait_asynccnt(0);
#else
  for (int i = threadIdx.x; i < count; i += blockDim.x) sdst[i] = gsrc[i];
#endif
}

// ---------------------------------------------------------------------------
// Kernel 0: zero workspace state that is accumulated into.
// ---------------------------------------------------------------------------
__global__ void k_init(unsigned long long* bestKey, int nKey,
                       int* numPos, int* numPosL, int N,
                       float* partL, float* partM, float* partC, int nPad) {
  int i = blockIdx.x * blockDim.x + threadIdx.x;
  int stride = gridDim.x * blockDim.x;
  for (int j = i; j < nKey; j += stride) bestKey[j] = 0ull;
  for (int j = i; j < N; j += stride) { numPos[j] = 0; numPosL[j] = 0; }
  for (int j = i; j < nPad; j += stride) { partL[j] = 0.f; partM[j] = 0.f; partC[j] = 0.f; }
}

// ---------------------------------------------------------------------------
// Kernel 1: per-truth argmax over priors (best_prior_idx), via u64 atomicMax.
// key = (bits(iou) << 32) | ~prior_idx  -> max picks highest iou, then lowest
// prior index on ties (matches jnp.argmax first-occurrence).
// ---------------------------------------------------------------------------
__global__ void k_match(const float4* __restrict__ priors,
                        const float* __restrict__ targets,
                        unsigned long long* __restrict__ bestKey,
                        int T, int P) {
  const int n = blockIdx.y;
  const int p = blockIdx.x * blockDim.x + threadIdx.x;
  __shared__ float sT[TMAX * 15];
  __shared__ unsigned long long sKey[TMAX];

  stage_to_lds(targets + (size_t)n * T * 15, sT, T * 15);
  for (int i = threadIdx.x; i < T; i += blockDim.x) sKey[i] = 0ull;
  __syncthreads();

  if (p < P) {
    float4 pr = priors[p];
    float px1 = pr.x - pr.z * 0.5f, py1 = pr.y - pr.w * 0.5f;
    float px2 = pr.x + pr.z * 0.5f, py2 = pr.y + pr.w * 0.5f;
    float area_b = (px2 - px1) * (py2 - py1);
    unsigned keyLo = ~(unsigned)p;
    for (int j = 0; j < T; ++j) {
      const float* t = &sT[j * 15];
      float ix = fminf(t[2], px2) - fmaxf(t[0], px1); ix = fmaxf(ix, 0.f);
      float iy = fminf(t[3], py2) - fmaxf(t[1], py1); iy = fmaxf(iy, 0.f);
      float inter = ix * iy;
      float area_a = (t[2] - t[0]) * (t[3] - t[1]);
      float ov = inter / (area_a + area_b - inter);  // iou in [0,1]
      unsigned long long key = ((unsigned long long)__float_as_uint(ov) << 32) | keyLo;
      atomicMax(&sKey[j], key);
    }
  }
  __syncthreads();
  for (int i = threadIdx.x; i < T; i += blockDim.x)
    atomicMax(&bestKey[(size_t)n * T + i], sKey[i]);
}

// ---------------------------------------------------------------------------
// Kernel 2: per-prior matching + all loss terms. Writes loss_rank for the
// hard-negative top-k, per-block fixed-order partial sums, integer counts.
// ---------------------------------------------------------------------------
__global__ void k_loss(const float4* __restrict__ loc4,
                       const float2* __restrict__ conf2,
                       const float2* __restrict__ lmd2,
                       const float4* __restrict__ priors,
                       const float* __restrict__ targets,
                       const unsigned long long* __restrict__ bestKey,
                       float* __restrict__ lossRank,
                       float* __restrict__ partL, float* __restrict__ partM,
                       float* __restrict__ partC,
                       int* __restrict__ numPos, int* __restrict__ numPosL,
                       int T, int P, int blocksPerBatch) {
  const int n = blockIdx.y;
  const int p = blockIdx.x * blockDim.x + threadIdx.x;
  const int tid = threadIdx.x;
  __shared__ float sT[TMAX * 15];
  __shared__ int sBPI[TMAX];
  __shared__ float red[256];
  __shared__ int sCnt[2];

  stage_to_lds(targets + (size_t)n * T * 15, sT, T * 15);
  for (int i = tid; i < T; i += blockDim.x)
    sBPI[i] = (int)(~(unsigned)(bestKey[(size_t)n * T + i] & 0xFFFFFFFFull));
  if (tid == 0) { sCnt[0] = 0; sCnt[1] = 0; }
  __syncthreads();

  float ll = 0.f, lmsum = 0.f, cep = 0.f;
  int posi = 0, posli = 0;
  if (p < P) {
    float4 pr = priors[p];
    float pcx = pr.x, pcy = pr.y, pw = pr.z, ph = pr.w;
    float px1 = pcx - pw * 0.5f, py1 = pcy - ph * 0.5f;
    float px2 = pcx + pw * 0.5f, py2 = pcy + ph * 0.5f;
    float area_b = (px2 - px1) * (py2 - py1);

    // per-prior best truth (argmax over T, first-occurrence on ties)
    float best_ov = -1.f; int best_j = 0;
    for (int j = 0; j < T; ++j) {
      const float* t = &sT[j * 15];
      float ix = fminf(t[2], px2) - fmaxf(t[0], px1); ix = fmaxf(ix, 0.f);
      float iy = fminf(t[3], py2) - fmaxf(t[1], py1); iy = fmaxf(iy, 0.f);
      float inter = ix * iy;
      float area_a = (t[2] - t[0]) * (t[3] - t[1]);
      float ov = inter / (area_a + area_b - inter);
      if (ov > best_ov) { best_ov = ov; best_j = j; }
    }
    // forced match override: overlap := 2.0, idx := last matching j
    bool forced = false;
    for (int j = 0; j < T; ++j) if (sBPI[j] == p) { forced = true; best_j = j; }
    if (forced) best_ov = 2.0f;

    const float* t = &sT[best_j * 15];
    float label = t[14];
    int conf_t = (best_ov < 0.35f) ? 0 : (int)label;
    bool pos  = (conf_t != 0);
    bool posl = (conf_t > 0);
    posi  = pos  ? 1 : 0;
    posli = posl ? 1 : 0;
    size_t idx = (size_t)n * P + p;

    if (pos) {  // localization loss (masked, so only compute when needed)
      float4 ld = loc4[idx];
      float l0 = ((t[0] + t[2]) * 0.5f - pcx) / (0.1f * pw);
      float l1 = ((t[1] + t[3]) * 0.5f - pcy) / (0.1f * ph);
      float l2 = logf((t[2] - t[0]) / pw) / 0.2f;
      float l3 = logf((t[3] - t[1]) / ph) / 0.2f;
      ll = smooth_l1(ld.x - l0) + smooth_l1(ld.y - l1) +
           smooth_l1(ld.z - l2) + smooth_l1(ld.w - l3);
    }
    if (posl) {  // landmark loss
#pragma unroll
      for (int q = 0; q < 5; ++q) {
        float2 d = lmd2[idx * 5 + q];
        float g0 = (t[4 + 2 * q] - pcx) / (0.1f * pw);
        float g1 = (t[5 + 2 * q] - pcy) / (0.1f * ph);
        lmsum += smooth_l1(d.x - g0) + smooth_l1(d.y - g1);
      }
    }
    // cross entropy (2 classes): ce = logsumexp - logit[conf_t']
    float2 cd = conf2[idx];
    float a = cd.x, b = cd.y;
    float mx = fmaxf(a, b);
    float lse = mx + logf(expf(a - mx) + expf(b - mx));
    float ce = lse - (pos ? b : a);
    lossRank[idx] = pos ? 0.f : ce;  // ce >= 0 always
    if (pos) cep = ce;
  }

  atomicAdd(&sCnt[0], posi);
  atomicAdd(&sCnt[1], posli);

  // fixed-order tree reductions (deterministic)
  red[tid] = ll; __syncthreads();
  for (int s = blockDim.x / 2; s > 0; s >>= 1) { if (tid < s) red[tid] += red[tid + s]; __syncthreads(); }
  float sumL = red[0]; __syncthreads();
  red[tid] = lmsum; __syncthreads();
  for (int s = blockDim.x / 2; s > 0; s >>= 1) { if (tid < s) red[tid] += red[tid + s]; __syncthreads(); }
  float sumM = red[0]; __syncthreads();
  red[tid] = cep; __syncthreads();
  for (int s = blockDim.x / 2; s > 0; s >>= 1) { if (tid < s) red[tid] += red[tid + s]; __syncthreads(); }
  float sumC = red[0];

  if (tid == 0) {
    int bi = n * blocksPerBatch + blockIdx.x;
    partL[bi] = sumL; partM[bi] = sumM; partC[bi] = sumC;
    atomicAdd(&numPos[n],  sCnt[0]);
    atomicAdd(&numPosL[n], sCnt[1]);
  }
}

// ---------------------------------------------------------------------------
// Kernel 3: per-batch exact top-k sum of lossRank via byte radix-select on
// the (non-negative, hence monotone) float bit patterns.
// ---------------------------------------------------------------------------
__global__ void k_topk(const float* __restrict__ lossRank,
                       const int* __restrict__ numPos,
                       float* __restrict__ topkSum, int P) {
  const int n = blockIdx.x;
  const int t = threadIdx.x;
  const int NT = blockDim.x;
  const float* lr = lossRank + (size_t)n * P;
  __shared__ unsigned hist[256];
  __shared__ unsigned sPrefix;
  __shared__ int sRem;
  __shared__ float red[1024];

  int np = numPos[n];
  long long kk = 7LL * np;                       // NEGPOS * num_pos
  int k = (kk > (long long)(P - 1)) ? (P - 1) : (int)kk;
  if (k <= 0) { if (t == 0) topkSum[n] = 0.f; return; }
  if (t == 0) { sPrefix = 0u; sRem = k; }

  for (int byte = 3; byte >= 0; --byte) {
    for (int i = t; i < 256; i += NT) hist[i] = 0u;
    __syncthreads();
    unsigned prefix = sPrefix;
    unsigned hm = (byte == 3) ? 0u : (0xFFFFFFFFu << ((byte + 1) * 8));
    for (int i = t; i < P; i += NT) {
      unsigned v = __float_as_uint(lr[i]);
      if ((v & hm) == (prefix & hm)) atomicAdd(&hist[(v >> (8 * byte)) & 255u], 1u);
    }
    __syncthreads();
    if (t == 0) {
      int rem = sRem;
      unsigned cum = 0;
      int b = 255;
      for (; b >= 0; --b) {
        unsigned h = hist[b];
        if (cum + h >= (unsigned)rem) break;
        cum += h;
      }
      if (b < 0) b = 0;
      sPrefix = prefix | ((unsigned)b << (8 * byte));
      sRem = rem - (int)cum;
    }
    __syncthreads();
  }
  unsigned thresh = sPrefix;
  int rem = sRem;  // multiplicity of threshold value to include

  float s = 0.f;
  for (int i = t; i < P; i += NT) {
    float v = lr[i];
    if (__float_as_uint(v) > thresh) s += v;
  }
  red[t] = s; __syncthreads();
  for (int st = NT / 2; st > 0; st >>= 1) { if (t < st) red[t] += red[t + st]; __syncthreads(); }
  if (t == 0) topkSum[n] = red[0] + (float)rem * __uint_as_float(thresh);
}

// ---------------------------------------------------------------------------
// Kernel 4: final reduction of block partials + loss assembly.
// Uses v_wmma_f32_16x16x4_f32 with an all-ones A matrix as an exact f32
// matrix-engine reduction: D = 1(16x4) x B(4x16) + C accumulates column sums;
// summing all 16 columns yields the total regardless of B's (k,n) layout.
// One wave (32 threads), EXEC all ones, uniform control flow.
// ---------------------------------------------------------------------------
__global__ void k_final(const float* __restrict__ partL,
                        const float* __restrict__ partM,
                        const float* __restrict__ partC,
                        int nPad,
                        const float* __restrict__ topk,
                        const int* __restrict__ numPos,
                        const int* __restrict__ numPosL,
                        int N, float* __restrict__ out) {
  const int lane = threadIdx.x;  // 32 threads
  float sums[3];
#if USE_WMMA_REDUCE
  v2f A; A.x = 1.0f; A.y = 1.0f;  // all-ones 16x4 A matrix
#pragma unroll
  for (int q = 0; q < 3; ++q) {
    const float* pq = (q == 0) ? partL : ((q == 1) ? partM : partC);
    v8f acc = {0.f, 0.f, 0.f, 0.f, 0.f, 0.f, 0.f, 0.f};
    for (int base = 0; base < nPad; base += 64) {  // 64 values per WMMA
      v2f B;
      B.x = pq[base + lane];
      B.y = pq[base + 32 + lane];
      acc = __builtin_amdgcn_wmma_f32_16x16x4_f32(
          /*neg_a=*/false, A, /*neg_b=*/false, B,
          /*c_mod=*/(short)0, acc, /*reuse_a=*/false, /*reuse_b=*/false);
    }
    // every row of D holds the column sums; take row 0 element from lanes 0..15
    float c = acc[0];
    float tot = 0.f;
    for (int i = 0; i < 16; ++i) tot += __shfl(c, i, 32);  // fixed order
    sums[q] = tot;
  }
#else
#pragma unroll
  for (int q = 0; q < 3; ++q) {
    const float* pq = (q == 0) ? partL : ((q == 1) ? partM : partC);
    float s = 0.f;
    for (int i = lane; i < nPad; i += 32) s += pq[i];
    float tot = 0.f;
    for (int i = 0; i < 32; ++i) tot += __shfl(s, i, 32);
    sums[q] = tot;
  }
#endif
  if (lane == 0) {
    float tk = 0.f;
    int np = 0, npl = 0;
    for (int n = 0; n < N; ++n) { tk += topk[n]; np += numPos[n]; npl += numPosL[n]; }
    float Nf  = fmaxf((float)np, 1.0f);
    float N1f = fmaxf((float)npl, 1.0f);
    out[0] = sums[0] / Nf;         // loss_l / N
    out[1] = (sums[2] + tk) / Nf;  // loss_c / N  (pos-ce + top-k negative ce)
    out[2] = sums[1] / N1f;        // loss_landm / N1
  }
}

// ---------------------------------------------------------------------------
extern "C" void kernel_launch(void* const* d_in, const int* in_sizes, int n_in,
                              void* d_out, int out_size, void* d_ws, size_t ws_size,
                              hipStream_t stream) {
  const float* loc     = (const float*)d_in[0];
  const float* conf    = (const float*)d_in[1];
  const float* landm   = (const float*)d_in[2];
  const float* priors  = (const float*)d_in[3];
  const float* targets = (const float*)d_in[4];

  const int P = in_sizes[3] / 4;              // 65536
  const int N = in_sizes[0] / (P * 4);        // 16
  const int T = in_sizes[4] / (N * 15);       // 48

  const int threads = 256;
  const int bpb  = (P + threads - 1) / threads;   // blocks per batch
  const int nblk = N * bpb;
  const int nPad = ((nblk + 63) / 64) * 64;       // multiple of 64 for WMMA tiles

  char* ws = (char*)d_ws;
  size_t off = 0;
  auto carve = [&](size_t bytes, size_t align) -> void* {
    off = (off + align - 1) & ~(align - 1);
    void* ptr = ws + off;
    off += bytes;
    return ptr;
  };
  unsigned long long* bestKey = (unsigned long long*)carve((size_t)N * T * 8, 8);
  int*   numPos   = (int*)carve((size_t)N * 4, 4);
  int*   numPosL  = (int*)carve((size_t)N * 4, 4);
  float* topk     = (float*)carve((size_t)N * 4, 4);
  float* partL    = (float*)carve((size_t)nPad * 4, 16);
  float* partM    = (float*)carve((size_t)nPad * 4, 16);
  float* partC    = (float*)carve((size_t)nPad * 4, 16);
  float* lossRank = (float*)carve((size_t)N * P * 4, 16);
  (void)ws_size; (void)n_in; (void)out_size;

  k_init<<<64, 256, 0, stream>>>(bestKey, N * T, numPos, numPosL, N,
                                 partL, partM, partC, nPad);
  dim3 grid(bpb, N);
  k_match<<<grid, threads, 0, stream>>>((const float4*)priors, targets, bestKey, T, P);
  k_loss<<<grid, threads, 0, stream>>>((const float4*)loc, (const float2*)conf,
                                       (const float2*)landm, (const float4*)priors,
                                       targets, bestKey, lossRank,
                                       partL, partM, partC, numPos, numPosL,
                                       T, P, bpb);
  k_topk<<<N, 1024, 0, stream>>>(lossRank, numPos, topk, P);
  k_final<<<1, 32, 0, stream>>>(partL, partM, partC, nPad, topk,
                                numPos, numPosL, N, (float*)d_out);
}